// FeatureTokenizer_22548578304376
// MI455X (gfx1250) — compile-verified
//
#include <hip/hip_runtime.h>
#include <hip/hip_bf16.h>

// CDNA5 / gfx1250 feature tokenizer.
// out[b, 0:64, d]  = x_num[b,f] * W_num[f,d] + b_num[f,d]   (via V_WMMA_F32_16X16X4_F32)
// out[b, 64:96, d] = emb_tables[f, x_cat[b,f], d]           (coalesced float4 gather)
//
// Memory-bound: ~403 MB stores + ~142 MB reads (emb table is L2-resident).

typedef __attribute__((ext_vector_type(2))) float v2f;
typedef __attribute__((ext_vector_type(8))) float v8f;

#define BATCH      16384
#define N_NUM      64
#define N_CAT      32
#define CARD       1000
#define DIM        64
#define TOKENS     (N_NUM + N_CAT)          // 96
#define ROW_STRIDE (TOKENS * DIM)           // 6144 floats per batch row

// ---------------------------------------------------------------------------
// Kernel 1: numeric tokens via WMMA f32 16x16x4.
// One wave per (batch-tile of 16, feature f); loops 4 d-tiles of 16.
//   A[m][k] : k==0 -> x_num[b0+m, f], else 0   (VGPR0 lanes 0-15 hold K=0)
//   B[k][n] : k==0 -> W_num[f, d0+n],  else 0  (VGPR0 lanes 0-15 hold K=0)
//   C[m][n] = b_num[f, d0+n]  (same value in all 8 accumulator VGPRs per lane)
//   D[m][n] = x_num[b0+m,f]*W_num[f,d0+n] + b_num[f,d0+n]
// ---------------------------------------------------------------------------
__global__ __launch_bounds__(256) void num_tokens_wmma_kernel(
    const float* __restrict__ x_num,   // [B, 64]
    const float* __restrict__ W_num,   // [64, 64]
    const float* __restrict__ b_num,   // [64, 64]
    float* __restrict__ out)           // [B, 96, 64]
{
    const int lane = threadIdx.x & 31;
    const int wave = threadIdx.x >> 5;                 // 8 waves / block
    const int gw   = blockIdx.x * 8 + wave;            // 65536 waves total
    const int f    = gw & (N_NUM - 1);                 // feature 0..63 (fast)
    const int bt   = gw >> 6;                          // batch tile 0..1023
    const int b0   = bt << 4;

    const int  l15 = lane & 15;
    const bool lo  = lane < 16;
    const int  rowAdd = lo ? 0 : 8;                    // C/D: lanes 16-31 hold M+8

    // A-matrix: only K=0 column non-zero (lanes 0-15 of VGPR0).
    const float xv = x_num[(b0 + l15) * N_NUM + f];
    v2f A;
    A.x = lo ? xv : 0.0f;                              // K=0 (lanes 0-15); K=2 -> 0
    A.y = 0.0f;                                        // K=1 / K=3 -> 0

    #pragma unroll
    for (int dt = 0; dt < 4; ++dt) {
        const int d0 = dt << 4;

        const float wv = W_num[f * DIM + d0 + l15];
        const float cv = b_num[f * DIM + d0 + l15];

        // B-matrix: only K=0 row non-zero (lanes 0-15 of VGPR0).
        v2f Bm;
        Bm.x = lo ? wv : 0.0f;
        Bm.y = 0.0f;

        // C: b_num broadcast across the 16 batch rows of the tile.
        v8f C;
        #pragma unroll
        for (int v = 0; v < 8; ++v) C[v] = cv;

        // 8 args: (neg_a, A, neg_b, B, c_mod, C, reuse_a, reuse_b)
        v8f D = __builtin_amdgcn_wmma_f32_16x16x4_f32(
            /*neg_a=*/false, A, /*neg_b=*/false, Bm,
            /*c_mod=*/(short)0, C, /*reuse_a=*/false, /*reuse_b=*/false);

        // D: VGPR v, lane l -> batch row b0 + v + rowAdd, column d0 + (l&15).
        size_t base = (size_t)(b0 + rowAdd) * ROW_STRIDE + (size_t)f * DIM + d0 + l15;
        #pragma unroll
        for (int v = 0; v < 8; ++v)
            out[base + (size_t)v * ROW_STRIDE] = D[v];
    }
}

// ---------------------------------------------------------------------------
// Kernel 2: categorical tokens — embedding row gather.
// 16 threads x float4 = one 256 B embedding row per (b, f) pair.
// Table is 8 MB -> lives in the 192 MB L2 after first touch.
// ---------------------------------------------------------------------------
__global__ __launch_bounds__(256) void cat_tokens_gather_kernel(
    const int*   __restrict__ x_cat,   // [B, 32]
    const float* __restrict__ emb,     // [32, 1000, 64]
    float* __restrict__ out)           // [B, 96, 64]
{
    const int gid  = blockIdx.x * 256 + threadIdx.x;   // B*N_CAT*16 threads
    const int q    = gid & 15;                         // float4 slot within row
    const int pair = gid >> 4;
    const int f    = pair & (N_CAT - 1);
    const int b    = pair >> 5;

    const int idx = x_cat[(size_t)b * N_CAT + f];

    const float4 v = *(const float4*)(emb + ((size_t)f * CARD + (size_t)idx) * DIM + q * 4);
    *(float4*)(out + (size_t)b * ROW_STRIDE + (size_t)(N_NUM + f) * DIM + q * 4) = v;
}

extern "C" void kernel_launch(void* const* d_in, const int* in_sizes, int n_in,
                              void* d_out, int out_size, void* d_ws, size_t ws_size,
                              hipStream_t stream) {
    const float* x_num = (const float*)d_in[0];
    const int*   x_cat = (const int*)  d_in[1];
    const float* W_num = (const float*)d_in[2];
    const float* b_num = (const float*)d_in[3];
    const float* emb   = (const float*)d_in[4];
    float* out = (float*)d_out;

    // 1024 batch-tiles * 64 features = 65536 waves / 8 waves per block.
    num_tokens_wmma_kernel<<<8192, 256, 0, stream>>>(x_num, W_num, b_num, out);

    // 16384 * 32 * 16 threads / 256.
    cat_tokens_gather_kernel<<<32768, 256, 0, stream>>>(x_cat, emb, out);
}